// TH3_38783554683569
// MI455X (gfx1250) — compile-verified
//
#include <hip/hip_runtime.h>

// out = x + x * sigmoid(2 * x * mask1), elementwise over 64*256*64*64 fp32.
// Memory-bound: 12 B/element, ~805 MB total -> ~35 us floor at 23.3 TB/s.
// Strategy: B128 non-temporal loads/stores, grid-stride, hardware exp/rcp.

typedef float v4f __attribute__((ext_vector_type(4)));

__device__ __forceinline__ float gated(float x, float m) {
    // t = 2*x*m ; sigmoid(t) = 1 / (1 + exp2(-t * log2(e)))
    const float LOG2E_X2 = 2.8853900817779268f;  // 2 * log2(e)
    float t  = x * m;
    float e  = __builtin_amdgcn_exp2f(-t * LOG2E_X2);   // v_exp_f32
    float g  = __builtin_amdgcn_rcpf(1.0f + e);         // v_rcp_f32
    return __builtin_fmaf(x, g, x);                     // x + x*g
}

__global__ __launch_bounds__(256) void TH3_gate_kernel(
    const float* __restrict__ x,
    const float* __restrict__ m,
    float* __restrict__ out,
    long long n)                       // total scalar elements
{
    const long long tid     = (long long)blockIdx.x * blockDim.x + threadIdx.x;
    const long long nthread = (long long)gridDim.x * blockDim.x;
    const long long n4      = n >> 2;  // float4 count

    const v4f* __restrict__ x4 = (const v4f*)x;
    const v4f* __restrict__ m4 = (const v4f*)m;
    v4f*       __restrict__ o4 = (v4f*)out;

    // Main vectorized stream: 16 B/lane loads & stores, non-temporal
    // (streaming data, no reuse -> don't pollute L2).
    for (long long i = tid; i < n4; i += nthread) {
        v4f xv = __builtin_nontemporal_load(&x4[i]);   // global_load_b128 NT
        v4f mv = __builtin_nontemporal_load(&m4[i]);
        v4f ov;
        ov.x = gated(xv.x, mv.x);
        ov.y = gated(xv.y, mv.y);
        ov.z = gated(xv.z, mv.z);
        ov.w = gated(xv.w, mv.w);
        __builtin_nontemporal_store(ov, &o4[i]);       // global_store_b128 NT
    }

    // Scalar tail (N here is a power of two, but stay generic/deterministic).
    for (long long k = (n4 << 2) + tid; k < n; k += nthread) {
        float xv = x[k];
        float mv = m[k];
        out[k] = gated(xv, mv);
    }
}

extern "C" void kernel_launch(void* const* d_in, const int* in_sizes, int n_in,
                              void* d_out, int out_size, void* d_ws, size_t ws_size,
                              hipStream_t stream) {
    const float* x = (const float*)d_in[0];
    const float* m = (const float*)d_in[1];
    float* out     = (float*)d_out;

    const long long n = (long long)in_sizes[0];   // 67,108,864

    // ~8 float4s per thread: n/4/8/256 blocks; 256 threads = 8 wave32 waves.
    const int block = 256;
    long long want = (n / 4 + (long long)block * 8 - 1) / ((long long)block * 8);
    int grid = (int)(want < 1 ? 1 : (want > 1048576 ? 1048576 : want));

    TH3_gate_kernel<<<grid, block, 0, stream>>>(x, m, out, n);
}